// PureGCN_34720515620917
// MI455X (gfx1250) — compile-verified
//
#include <hip/hip_runtime.h>
#include <hip/hip_bf16.h>

#define N_NODES 100000
#define D_FEAT  128
#define E_EDGES 1600000
#define LN_EPS  1e-5f

typedef __attribute__((ext_vector_type(2))) float v2f;
typedef __attribute__((ext_vector_type(8))) float v8f;

// ---------------------------------------------------------------------------
// Dense GEMM: H[n,h] = sum_k X[n,k] * W[h,k] + b[h]
// One wave per 16x16 output tile, fp32 WMMA (V_WMMA_F32_16X16X4_F32).
// A-frag (16x4): lane<16 -> (M=lane, K=0,1); lane>=16 -> (M=lane-16, K=2,3)
// B-frag (4x16): same pattern with N=lane&15 (B[k][n] = W[n][k], symmetric)
// D (16x16 f32): VGPR v -> row v (lanes 0-15) / row v+8 (lanes 16-31)
// ---------------------------------------------------------------------------
__global__ __launch_bounds__(256) void gemm_wmma_f32(
    const float* __restrict__ X, const float* __restrict__ W,
    const float* __restrict__ bias, float* __restrict__ H) {
  const int wave = (int)((blockIdx.x * blockDim.x + threadIdx.x) >> 5);
  const int lane = (int)(threadIdx.x & 31u);
  const int tilesN = D_FEAT / 16;                 // 8
  const int tileM = (wave / tilesN) * 16;         // node-row base
  const int tileN = (wave % tilesN) * 16;         // feature-col base
  if (tileM >= N_NODES) return;                   // wave-uniform exit

  const int halfLane = lane & 15;
  const int kOff = (lane >> 4) * 2;               // 0 or 2
  const float* __restrict__ xrow = X + (size_t)(tileM + halfLane) * D_FEAT;
  const float* __restrict__ wrow = W + (size_t)(tileN + halfLane) * D_FEAT;

  v8f acc = {};
#pragma unroll 4
  for (int k = 0; k < D_FEAT; k += 4) {
    v2f a, b;
    a.x = xrow[k + kOff];
    a.y = xrow[k + kOff + 1];
    b.x = wrow[k + kOff];
    b.y = wrow[k + kOff + 1];
    acc = __builtin_amdgcn_wmma_f32_16x16x4_f32(
        /*neg_a=*/false, a, /*neg_b=*/false, b,
        /*c_mod=*/(short)0, acc, /*reuse_a=*/false, /*reuse_b=*/false);
  }

  const float bv = bias[tileN + halfLane];
  const int colC = tileN + halfLane;
  const int rowBase = tileM + ((lane >> 4) << 3); // +8 for upper half-wave
#pragma unroll
  for (int v = 0; v < 8; ++v)
    H[(size_t)(rowBase + v) * D_FEAT + colC] = acc[v] + bv;
}

// ---------------------------------------------------------------------------
// Optional residual add + LayerNorm + ReLU. One wave per node (wave32:
// 4 features per lane), reductions via __shfl_xor.
// ---------------------------------------------------------------------------
__global__ __launch_bounds__(256) void ln_relu(
    const float* __restrict__ Hin, const float* __restrict__ Res,
    const float* __restrict__ scale, const float* __restrict__ bias,
    float* __restrict__ T, int addRes) {
  const int node = (int)((blockIdx.x * blockDim.x + threadIdx.x) >> 5);
  const int lane = (int)(threadIdx.x & 31u);
  if (node >= N_NODES) return;

  const size_t base = (size_t)node * D_FEAT;
  float v[4];
#pragma unroll
  for (int j = 0; j < 4; ++j) {
    const int c = lane + 32 * j;
    v[j] = Hin[base + c];
    if (addRes) v[j] += Res[base + c];
  }
  float s = v[0] + v[1] + v[2] + v[3];
#pragma unroll
  for (int off = 16; off > 0; off >>= 1) s += __shfl_xor(s, off, 32);
  const float mu = s * (1.0f / (float)D_FEAT);

  float sq = 0.0f;
#pragma unroll
  for (int j = 0; j < 4; ++j) {
    const float d = v[j] - mu;
    sq += d * d;
  }
#pragma unroll
  for (int off = 16; off > 0; off >>= 1) sq += __shfl_xor(sq, off, 32);
  const float rstd = rsqrtf(sq * (1.0f / (float)D_FEAT) + LN_EPS);

#pragma unroll
  for (int j = 0; j < 4; ++j) {
    const int c = lane + 32 * j;
    float o = (v[j] - mu) * rstd * scale[c] + bias[c];
    T[base + c] = o > 0.0f ? o : 0.0f;
  }
}

// ---------------------------------------------------------------------------
// Weighted SpMM scatter: Agg[dst] += T[src] * w.  One wave per edge;
// float4 (b128) gather per lane, 4x global_atomic_add_f32 scatter.
// Both T (51.2 MB) and Agg (51.2 MB) are L2-resident (192 MB L2).
// ---------------------------------------------------------------------------
__global__ __launch_bounds__(256) void spmm_scatter(
    const float* __restrict__ T, const long long* __restrict__ src,
    const long long* __restrict__ dst, const float* __restrict__ ef,
    float* __restrict__ Agg) {
  const int e = (int)((blockIdx.x * blockDim.x + threadIdx.x) >> 5);
  const int lane = (int)(threadIdx.x & 31u);
  if (e >= E_EDGES) return;

  const int s = (int)src[e];
  const int d = (int)dst[e];
  const float w = ef[e];

  const float4 m = *(const float4*)(T + (size_t)s * D_FEAT + lane * 4);
  float* __restrict__ ap = Agg + (size_t)d * D_FEAT + lane * 4;
  unsafeAtomicAdd(ap + 0, m.x * w);
  unsafeAtomicAdd(ap + 1, m.y * w);
  unsafeAtomicAdd(ap + 2, m.z * w);
  unsafeAtomicAdd(ap + 3, m.w * w);
}

__global__ __launch_bounds__(256) void zero_f32(float* __restrict__ p, size_t n) {
  const size_t i = ((size_t)blockIdx.x * blockDim.x + threadIdx.x) * 4;
  if (i < n) *(float4*)(p + i) = make_float4(0.f, 0.f, 0.f, 0.f);
}

extern "C" void kernel_launch(void* const* d_in, const int* in_sizes, int n_in,
                              void* d_out, int out_size, void* d_ws, size_t ws_size,
                              hipStream_t stream) {
  const float*      x    = (const float*)d_in[0];
  const long long*  eidx = (const long long*)d_in[1]; // [2, E] int64
  const float*      ef   = (const float*)d_in[2];
  const float*      lw   = (const float*)d_in[3];     // [HID, D_IN]
  const float*      lb   = (const float*)d_in[4];     // [HID]
  const float*      lns  = (const float*)d_in[5];     // [L, HID]
  const float*      lnb  = (const float*)d_in[6];     // [L, HID]
  float*            out  = (float*)d_out;             // [N, HID]

  const size_t nd = (size_t)N_NODES * D_FEAT;
  float* h0 = (float*)d_ws;        // GEMM result (kept for residual)
  float* t  = h0 + nd;             // post LN/ReLU features
  float* h1 = t + nd;              // layer-0 aggregate

  const long long* srcI = eidx;
  const long long* dstI = eidx + E_EDGES;

  const dim3 blk(256);
  const int gemmBlocks = ((N_NODES / 16) * (D_FEAT / 16) * 32 + 255) / 256; // 6250
  const int nodeBlocks = (N_NODES * 32 + 255) / 256;                        // 12500
  const int edgeBlocks = (int)(((size_t)E_EDGES * 32 + 255) / 256);         // 200000
  const int zeroBlocks = (int)((nd / 4 + 255) / 256);

  // h0 = x @ W^T + b   (fp32 WMMA)
  gemm_wmma_f32<<<gemmBlocks, blk, 0, stream>>>(x, lw, lb, h0);

  // Layer 0: t = relu(LN(h0)); h1 = scatter-add
  ln_relu<<<nodeBlocks, blk, 0, stream>>>(h0, nullptr, lns, lnb, t, 0);
  zero_f32<<<zeroBlocks, blk, 0, stream>>>(h1, nd);
  spmm_scatter<<<edgeBlocks, blk, 0, stream>>>(t, srcI, dstI, ef, h1);

  // Layer 1: t = relu(LN(h1 + h0)); out = scatter-add
  ln_relu<<<nodeBlocks, blk, 0, stream>>>(h1, h0, lns + D_FEAT, lnb + D_FEAT, t, 1);
  zero_f32<<<zeroBlocks, blk, 0, stream>>>(out, nd);
  spmm_scatter<<<edgeBlocks, blk, 0, stream>>>(t, srcI, dstI, ef, out);
}